// HMMTrajNet_21612275433732
// MI455X (gfx1250) — compile-verified
//
#include <hip/hip_runtime.h>

// HMM forward pass, MI455X (gfx1250, wave32).
// Roofline: ~350 MB streamed once => ~15us at 23.3 TB/s. Compute is trivial, so
// the design is (a) 4096-way parallel chunked associative scan (the log-space
// step is an associative log-matmul), (b) rank-1 + diagonal structure of the
// linear-space per-step transition (all factors in (0,1], no overflow),
// (c) fp32 WMMA (v_wmma_f32_16x16x4_f32) for the chunk-product combine,
// (d) native v_exp_f32 / v_log_f32 / v_rcp_f32 in the hot loop.

#define NSTATE   16
#define NACT     18
#define NBATCH   64
#define MAXT     4096
#define TP1      4097
#define NCHUNK   64
#define CHUNKLEN 64

typedef float v2f __attribute__((ext_vector_type(2)));
typedef float v8f __attribute__((ext_vector_type(8)));

__device__ __forceinline__ float fast_rcp(float x) {
  return __builtin_amdgcn_rcpf(x);  // v_rcp_f32
}

__device__ __forceinline__ float wave_max32(float v) {
#pragma unroll
  for (int off = 16; off >= 1; off >>= 1)
    v = fmaxf(v, __shfl_xor(v, off, 32));
  return v;
}

__device__ __forceinline__ float wave_sum32(float v) {
#pragma unroll
  for (int off = 16; off >= 1; off >>= 1)
    v += __shfl_xor(v, off, 32);
  return v;
}

// ---------------------------------------------------------------------------
// Kernel 1: per-(batch, chunk) fold of CHUNKLEN steps into one normalized
// linear-space 16x16 matrix + log-scale. One wave32 per workgroup.
// Lane l owns P[row = l&15][cols 8*(l>>4) .. +7] (8 fp32 registers).
// Per step i (only while i < length):
//   eb[j] = exp(stop[i,j,STOP]); w1[k] = exp(start[i,k]+al[k]);
//   w2[k] = exp(stop[i,k,CONT]+al[k]);  al[k] = action_logps[i,k,actions[i]]
//   P <- (P @ eb) (x) w1  +  P * diag(w2)
// Renormalize by wave-max every 4 steps; accumulate log-scale.
// ---------------------------------------------------------------------------
__global__ __launch_bounds__(32) void hmm_chunk_scan(
    const float* __restrict__ action_logps,
    const float* __restrict__ stop_logps,
    const float* __restrict__ start_logps,
    const int*   __restrict__ actions,
    const int*   __restrict__ lengths,
    float* __restrict__ wsP,
    float* __restrict__ wsScale)
{
  const int chunk = blockIdx.x;
  const int bb    = blockIdx.y;
  const int lane  = threadIdx.x;
  const int row   = lane & 15;
  const int h     = lane >> 4;      // which 8-column half this lane owns
  const int c0    = h * 8;

  __shared__ __align__(16) float eb_s[NSTATE];
  __shared__ __align__(16) float w1_s[NSTATE];
  __shared__ __align__(16) float w2_s[NSTATE];

  float p[8];
#pragma unroll
  for (int k = 0; k < 8; ++k) p[k] = ((c0 + k) == row) ? 1.0f : 0.0f;
  float scale = 0.0f;

  const int len  = lengths[bb];
  const int iBeg = 1 + chunk * CHUNKLEN;
  int iEnd = iBeg + CHUNKLEN;
  if (iEnd > MAXT) iEnd = MAXT;   // reference scan runs i = 1 .. MAXT-1
  if (iEnd > len)  iEnd = len;    // i >= length => identity step (skip)

  int sincescale = 0;
  for (int i = iBeg; i < iEnd; ++i) {
    if (h == 0) {
      const long long base = ((long long)bb * TP1 + i) * NSTATE + row;
      const float2 st2 = ((const float2*)stop_logps)[base];  // (beta, omb)
      const float  st  = start_logps[base];
      const int    act = actions[(long long)bb * MAXT + i];  // wave-uniform
      const float  al  = action_logps[base * NACT + act];
      eb_s[row] = __expf(st2.x);          // native v_exp_f32 path
      w1_s[row] = __expf(st + al);
      w2_s[row] = __expf(st2.y + al);
      if (i + 1 < iEnd) {  // prefetch next step's action block (global_prefetch_b8)
        const long long nb = ((long long)bb * TP1 + (i + 1)) * NSTATE + row;
        __builtin_prefetch(action_logps + nb * NACT, 0, 0);
      }
    }
    __syncthreads();

    const float4 ebA = ((const float4*)eb_s)[2 * h];
    const float4 ebB = ((const float4*)eb_s)[2 * h + 1];
    float partial = p[0] * ebA.x + p[1] * ebA.y + p[2] * ebA.z + p[3] * ebA.w
                  + p[4] * ebB.x + p[5] * ebB.y + p[6] * ebB.z + p[7] * ebB.w;
    // pair lane with lane^16 (same row, other column half) to finish the dot
    const float r = partial + __shfl_xor(partial, 16, 32);

    const float4 w1A = ((const float4*)w1_s)[2 * h];
    const float4 w1B = ((const float4*)w1_s)[2 * h + 1];
    const float4 w2A = ((const float4*)w2_s)[2 * h];
    const float4 w2B = ((const float4*)w2_s)[2 * h + 1];
    p[0] = fmaf(r, w1A.x, p[0] * w2A.x);
    p[1] = fmaf(r, w1A.y, p[1] * w2A.y);
    p[2] = fmaf(r, w1A.z, p[2] * w2A.z);
    p[3] = fmaf(r, w1A.w, p[3] * w2A.w);
    p[4] = fmaf(r, w1B.x, p[4] * w2B.x);
    p[5] = fmaf(r, w1B.y, p[5] * w2B.y);
    p[6] = fmaf(r, w1B.z, p[6] * w2B.z);
    p[7] = fmaf(r, w1B.w, p[7] * w2B.w);
    __syncthreads();

    if (++sincescale == 4) {
      sincescale = 0;
      float m = fmaxf(fmaxf(fmaxf(p[0], p[1]), fmaxf(p[2], p[3])),
                      fmaxf(fmaxf(p[4], p[5]), fmaxf(p[6], p[7])));
      m = wave_max32(m);
      scale += __logf(m);               // native v_log_f32 path
      const float inv = fast_rcp(m);    // v_rcp_f32
#pragma unroll
      for (int k = 0; k < 8; ++k) p[k] *= inv;
    }
  }

  // final normalization (if loop was empty, P = I, m = 1, scale += 0)
  {
    float m = fmaxf(fmaxf(fmaxf(p[0], p[1]), fmaxf(p[2], p[3])),
                    fmaxf(fmaxf(p[4], p[5]), fmaxf(p[6], p[7])));
    m = wave_max32(m);
    scale += __logf(m);
    const float inv = fast_rcp(m);
#pragma unroll
    for (int k = 0; k < 8; ++k) p[k] *= inv;
  }

  // store row-major 16x16 + scale
  float* outP = wsP + (((long long)bb * NCHUNK + chunk) << 8);
  float4 lo = {p[0], p[1], p[2], p[3]};
  float4 hi = {p[4], p[5], p[6], p[7]};
  ((float4*)(outP + row * 16 + c0))[0] = lo;
  ((float4*)(outP + row * 16 + c0 + 4))[0] = hi;
  if (lane == 0) wsScale[bb * NCHUNK + chunk] = scale;
}

// ---------------------------------------------------------------------------
// Kernel 2: one wave per batch. Chain the 64 chunk matrices in order with
// fp32 WMMA (4x v_wmma_f32_16x16x4_f32 per 16x16x16 product), renormalizing
// by the matrix max each multiply (log-scale accumulated in fp32). Then fold
// in f0 = start[0]+al0 and terminal stop logps and emit per-batch log p.
//
// ISA layouts (CDNA5 ISA):
//   A 16x4 fp32 : lanes 0-15 hold M=0..15 with K=0 (v0), K=1 (v1);
//                 lanes 16-31 hold K=2 (v0), K=3 (v1).
//   B 4x16 fp32 : mirrored — v0: K=2h row, v1: K=2h+1 row, N = lane&15.
//   C/D 16x16   : VGPR v holds M = v + 8*(lane>=16), N = lane&15.
// ---------------------------------------------------------------------------
__global__ __launch_bounds__(32) void hmm_combine(
    const float* __restrict__ wsP,
    const float* __restrict__ wsScale,
    const float* __restrict__ action_logps,
    const float* __restrict__ stop_logps,
    const float* __restrict__ start_logps,
    const int*   __restrict__ actions,
    const int*   __restrict__ lengths,
    float* __restrict__ wsLogp)
{
  const int bb   = blockIdx.x;
  const int lane = threadIdx.x;
  const int n    = lane & 15;
  const int h    = lane >> 4;

  __shared__ float L[16 * 17];  // running left product, row-major, pad 17
  __shared__ float fs[NSTATE];

  // L = chunk 0
  const float* P0 = wsP + ((long long)bb * NCHUNK << 8);
#pragma unroll
  for (int k = 0; k < 8; ++k) {
    const int idx = lane * 8 + k;
    L[(idx >> 4) * 17 + (idx & 15)] = P0[idx];
  }
  float s = wsScale[bb * NCHUNK];
  __syncthreads();

  for (int c = 1; c < NCHUNK; ++c) {
    const float* R = wsP + (((long long)bb * NCHUNK + c) << 8);
    s += wsScale[bb * NCHUNK + c];

    v8f acc = {0.f, 0.f, 0.f, 0.f, 0.f, 0.f, 0.f, 0.f};
#pragma unroll
    for (int k = 0; k < 4; ++k) {
      const int ka = 4 * k + 2 * h;
      v2f a, b;
      a.x = L[n * 17 + ka];        // A[M=n][K=ka]
      a.y = L[n * 17 + ka + 1];    // A[M=n][K=ka+1]
      b.x = R[ka * 16 + n];        // B[K=ka][N=n]
      b.y = R[(ka + 1) * 16 + n];  // B[K=ka+1][N=n]
      acc = __builtin_amdgcn_wmma_f32_16x16x4_f32(
          false, a, false, b, (short)0, acc, false, false);
    }

    // renormalize product by its max; fold into log-scale
    float m = acc[0];
#pragma unroll
    for (int v = 1; v < 8; ++v) m = fmaxf(m, acc[v]);
    m = wave_max32(m);
    s += __logf(m);
    const float inv = fast_rcp(m);
    __syncthreads();
#pragma unroll
    for (int v = 0; v < 8; ++v) L[(v + 8 * h) * 17 + n] = acc[v] * inv;
    __syncthreads();
  }

  // f0[j] = start_logps[bb,0,j] + action_logps[bb,0,j,actions[bb,0]]
  const int act0 = actions[(long long)bb * MAXT];
  float f0 = -3.0e38f;
  if (h == 0) {
    const long long base = (long long)bb * TP1 * NSTATE + n;
    f0 = start_logps[base] + action_logps[base * NACT + act0];
  }
  const float m0 = wave_max32(f0);
  if (h == 0) fs[n] = __expf(f0 - m0);
  __syncthreads();

  float term = 0.0f;
  if (h == 0) {
    float ff = 0.0f;
#pragma unroll
    for (int j = 0; j < NSTATE; ++j) ff += fs[j] * L[j * 17 + n];
    const int len = lengths[bb];
    const float stT =
        stop_logps[(((long long)bb * TP1 + len) * NSTATE + n) * 2 + 0];
    term = ff * __expf(stT);
  }
  const float tot = wave_sum32(term);
  if (lane == 0) wsLogp[bb] = __logf(tot) + s + m0;
}

// ---------------------------------------------------------------------------
// Kernel 3: reduce 64 per-batch log p's, write -sum.
// ---------------------------------------------------------------------------
__global__ __launch_bounds__(32) void hmm_final(const float* __restrict__ wsLogp,
                                                float* __restrict__ out)
{
  const int lane = threadIdx.x;
  float v = wsLogp[lane] + wsLogp[lane + 32];
  v = wave_sum32(v);
  if (lane == 0) out[0] = -v;
}

extern "C" void kernel_launch(void* const* d_in, const int* in_sizes, int n_in,
                              void* d_out, int out_size, void* d_ws, size_t ws_size,
                              hipStream_t stream) {
  (void)in_sizes; (void)n_in; (void)out_size; (void)ws_size;
  const float* action_logps = (const float*)d_in[0];
  const float* stop_logps   = (const float*)d_in[1];
  const float* start_logps  = (const float*)d_in[2];
  const int*   actions      = (const int*)d_in[3];
  const int*   lengths      = (const int*)d_in[4];
  float* out = (float*)d_out;

  // workspace: 64*64 matrices (256 f32) + 64*64 scales + 64 logps ~= 4.2 MB
  float* wsP     = (float*)d_ws;
  float* wsScale = wsP + (size_t)NBATCH * NCHUNK * 256;
  float* wsLogp  = wsScale + (size_t)NBATCH * NCHUNK;

  hmm_chunk_scan<<<dim3(NCHUNK, NBATCH), dim3(32), 0, stream>>>(
      action_logps, stop_logps, start_logps, actions, lengths, wsP, wsScale);
  hmm_combine<<<dim3(NBATCH), dim3(32), 0, stream>>>(
      wsP, wsScale, action_logps, stop_logps, start_logps, actions, lengths,
      wsLogp);
  hmm_final<<<dim3(1), dim3(32), 0, stream>>>(wsLogp, out);
}